// RoINet_10385230921768
// MI455X (gfx1250) — compile-verified
//
#include <hip/hip_runtime.h>
#include <stdint.h>

#define BN 8
#define PN 8000
#define CN 80
#define PCN (PN * CN)
#define KN 2048
#define KW 32            // KN/64 mask words per row
#define OUTN 100
#define F_SCORE_THRE 0.15f
#define F_NMS_THRESH 0.5f
#define F_MIN_AREA  0.1f

// NMS scan chunking: 32 rows/chunk -> 8KB/chunk, 16 async b128 ops/chunk
#define CH 32
#define NCH (KN / CH)            // 64
#define CHUNK_BYTES (CH * KW * 8)  // 8192
#define OPS_PER_CHUNK (CHUNK_BYTES / 512)  // 16 (512B per wave-op: 32 lanes x 16B)

// ---------------- workspace layout (bytes) ----------------
#define WS_SCORES 0ull                              // B*PC*4 = 20,480,000
#define WS_HIST   (WS_SCORES + 20480000ull)         // B*256*4
#define WS_PREFIX (WS_HIST   + 8192ull)             // B*8
#define WS_REMAIN (WS_PREFIX + 256ull)              // B*4
#define WS_SELCNT (WS_REMAIN + 256ull)              // B*4
#define WS_MAXC   (WS_SELCNT + 256ull)              // B*4
#define WS_KEYS   (WS_MAXC   + 256ull)              // B*K*8
#define WS_CB     (WS_KEYS   + 131072ull)           // B*K*4*4
#define WS_VALS   (WS_CB     + 524288ull)           // B*K*4
#define WS_LAB    (WS_VALS   + 65536ull)            // B*K*4
#define WS_MASK   (WS_LAB    + 65536ull)            // B*K*KW*8 = 4,194,304
#define WS_KEEP   (WS_MASK   + 4194304ull)          // B*KW*8

// ---------------- helpers ----------------
__device__ __forceinline__ unsigned int f2ord(float f) {
    unsigned int b = __float_as_uint(f);
    return (b & 0x80000000u) ? ~b : (b | 0x80000000u);
}
__device__ __forceinline__ float ord2f(unsigned int u) {
    unsigned int b = (u & 0x80000000u) ? (u & 0x7fffffffu) : ~u;
    return __uint_as_float(b);
}
__device__ __forceinline__ unsigned long long make_key(float f, unsigned int idx) {
    return ((unsigned long long)f2ord(f) << 32) | (unsigned long long)(unsigned int)(~idx);
}

// gfx1250 async global->LDS copies (ASYNCcnt path, CDNA5 ISA)
__device__ __forceinline__ void async_lds_b128(unsigned int lds_off, const void* gaddr) {
    asm volatile("global_load_async_to_lds_b128 %0, %1, off"
                 :: "v"(lds_off), "v"(gaddr) : "memory");
}
__device__ __forceinline__ void async_lds_b32(unsigned int lds_off, const void* gaddr) {
    asm volatile("global_load_async_to_lds_b32 %0, %1, off"
                 :: "v"(lds_off), "v"(gaddr) : "memory");
}
// flat pointer low 32 bits == wave-relative LDS address on gfx1250 (ISA §10.2 aperture map)
__device__ __forceinline__ unsigned int lds_addr(const void* p) {
    return (unsigned int)(uintptr_t)p;
}
// stage one 8KB chunk (CH rows x KW words) linearly into LDS; 16 async ops/wave
__device__ __forceinline__ void issue_chunk(void* dst, const void* src, int lane) {
    unsigned int l0 = lds_addr(dst) + (unsigned int)lane * 16u;
    const char* g = (const char*)src + lane * 16;
    #pragma unroll
    for (int k = 0; k < OPS_PER_CHUNK; ++k)
        async_lds_b128(l0 + (unsigned int)(k * 512), g + k * 512);
}

// ---------------- kernels ----------------
__global__ void k_init(unsigned int* hist, unsigned long long* prefix,
                       unsigned int* remain, unsigned int* selcnt, unsigned int* maxc) {
    int img = blockIdx.x, t = threadIdx.x;
    hist[img * 256 + t] = 0u;
    if (t == 0) {
        prefix[img] = 0ull;
        remain[img] = KN;
        selcnt[img] = 0u;
        maxc[img]   = f2ord(0.0f);  // 0x80000000
    }
}

// softmax + decode + validity -> masked scores (-1 invalid)
__global__ void k_scores(const float* __restrict__ logits,
                         const float* __restrict__ deltas,
                         const float* __restrict__ props,
                         float* __restrict__ scores) {
    int img = blockIdx.y;
    int p = blockIdx.x * blockDim.x + threadIdx.x;
    if (p >= PN) return;
    const float* L = logits + ((size_t)img * PN + p) * (CN + 1);
    float m = L[0];
    #pragma unroll 1
    for (int c = 1; c <= CN; ++c) m = fmaxf(m, L[c]);
    float s = 0.f;
    #pragma unroll 1
    for (int c = 0; c <= CN; ++c) s += expf(L[c] - m);
    float inv = 1.f / s;

    const float* pr = props + ((size_t)img * PN + p) * 4;
    float ax0 = pr[0], ay0 = pr[1], ax1 = pr[2], ay1 = pr[3];
    float wa = ax1 - ax0, ha = ay1 - ay0;
    float cxa = ax0 + 0.5f * wa, cya = ay0 + 0.5f * ha;

    const float* D = deltas + ((size_t)img * PN + p) * ((CN + 1) * 4);
    float* so = scores + (size_t)img * PCN + (size_t)p * CN;
    #pragma unroll 1
    for (int j = 0; j < CN; ++j) {
        float sc = expf(L[j + 1] - m) * inv;
        const float* d = D + 4 * (j + 1);
        float cx = d[0] * wa + cxa;
        float cy = d[1] * ha + cya;
        float w  = expf(d[2]) * wa;
        float h  = expf(d[3]) * ha;
        float x0 = cx - 0.5f * w, y0 = cy - 0.5f * h;
        float x1 = cx + 0.5f * w, y1 = cy + 0.5f * h;
        float area = (y1 - y0) * (x1 - x0);
        bool valid = (sc > F_SCORE_THRE) && (area > F_MIN_AREA);
        so[j] = valid ? sc : -1.0f;
    }
}

// radix-select pass: 256-bin histogram of keys matching current prefix
__global__ void k_hist(const float* __restrict__ scores,
                       const unsigned long long* __restrict__ prefix,
                       unsigned int* __restrict__ hist, int pass) {
    __shared__ unsigned int h[256];
    int t = threadIdx.x;
    if (t < 256) h[t] = 0u;
    __syncthreads();
    int img = blockIdx.y;
    int i = blockIdx.x * blockDim.x + t;
    int shift = 56 - 8 * pass;
    unsigned long long pf = prefix[img];
    if (i < PCN) {
        unsigned long long key = make_key(scores[(size_t)img * PCN + i], (unsigned int)i);
        bool ok = (pass == 0) || ((key >> (shift + 8)) == (pf >> (shift + 8)));
        if (ok) atomicAdd(&h[(unsigned int)(key >> shift) & 255u], 1u);
    }
    __syncthreads();
    if (t < 256 && h[t]) atomicAdd(&hist[img * 256 + t], h[t]);
}

__global__ void k_select(unsigned int* hist, unsigned long long* prefix,
                         unsigned int* remain, int pass) {
    __shared__ unsigned int h[256];
    int img = blockIdx.x, t = threadIdx.x;
    h[t] = hist[img * 256 + t];
    __syncthreads();
    hist[img * 256 + t] = 0u;      // ready for next pass
    if (t == 0) {
        unsigned int rem = remain[img];
        int sel = 0;
        for (int b = 255; b >= 0; --b) {
            unsigned int c = h[b];
            if (c >= rem) { sel = b; break; }
            rem -= c;
        }
        int shift = 56 - 8 * pass;
        prefix[img] |= ((unsigned long long)sel) << shift;
        remain[img] = rem;
    }
}

__global__ void k_compact(const float* __restrict__ scores,
                          const unsigned long long* __restrict__ prefix,
                          unsigned int* selcnt, unsigned long long* keys) {
    int img = blockIdx.y;
    int i = blockIdx.x * blockDim.x + threadIdx.x;
    if (i >= PCN) return;
    unsigned long long T = prefix[img];   // exact 2048-th largest key
    unsigned long long key = make_key(scores[(size_t)img * PCN + i], (unsigned int)i);
    if (key >= T) {
        unsigned int pos = atomicAdd(&selcnt[img], 1u);
        if (pos < KN) keys[(size_t)img * KN + pos] = key;
    }
}

// bitonic sort 2048 unique keys descending (LDS, 16KB)
__global__ void k_sort(unsigned long long* keys) {
    __shared__ unsigned long long sk[KN];
    int img = blockIdx.x, t = threadIdx.x;
    sk[t]        = keys[(size_t)img * KN + t];
    sk[t + 1024] = keys[(size_t)img * KN + t + 1024];
    for (int k = 2; k <= KN; k <<= 1) {
        for (int j = k >> 1; j > 0; j >>= 1) {
            __syncthreads();
            #pragma unroll
            for (int half = 0; half < 2; ++half) {
                int i = t + half * 1024;
                int l = i ^ j;
                if (l > i) {
                    bool desc = ((i & k) == 0);
                    unsigned long long a = sk[i], b = sk[l];
                    if (desc ? (a < b) : (a > b)) { sk[i] = b; sk[l] = a; }
                }
            }
        }
    }
    __syncthreads();
    keys[(size_t)img * KN + t]        = sk[t];
    keys[(size_t)img * KN + t + 1024] = sk[t + 1024];
}

// decode boxes for selected candidates, labels, vals, global max-coord
__global__ void k_decode(const unsigned long long* __restrict__ keys,
                         const float* __restrict__ deltas,
                         const float* __restrict__ props,
                         float* __restrict__ cb, float* __restrict__ vals,
                         int* __restrict__ labels, unsigned int* maxc) {
    __shared__ unsigned int sMax;
    int img = blockIdx.y;
    int k = blockIdx.x * blockDim.x + threadIdx.x;
    if (threadIdx.x == 0) sMax = f2ord(0.0f);
    __syncthreads();

    unsigned long long key = keys[(size_t)img * KN + k];
    float val = ord2f((unsigned int)(key >> 32));
    unsigned int idx = ~(unsigned int)(key & 0xffffffffull);
    int p = (int)(idx / CN), j = (int)(idx % CN);

    const float* pr = props + ((size_t)img * PN + p) * 4;
    float ax0 = pr[0], ay0 = pr[1], ax1 = pr[2], ay1 = pr[3];
    float wa = ax1 - ax0, ha = ay1 - ay0;
    float cxa = ax0 + 0.5f * wa, cya = ay0 + 0.5f * ha;
    const float* d = deltas + ((size_t)img * PN + p) * ((CN + 1) * 4) + 4 * (j + 1);
    float cx = d[0] * wa + cxa;
    float cy = d[1] * ha + cya;
    float w  = expf(d[2]) * wa;
    float h  = expf(d[3]) * ha;
    float x0 = cx - 0.5f * w, y0 = cy - 0.5f * h;
    float x1 = cx + 0.5f * w, y1 = cy + 0.5f * h;

    float* o = cb + ((size_t)img * KN + k) * 4;
    o[0] = x0; o[1] = y0; o[2] = x1; o[3] = y1;
    vals[(size_t)img * KN + k]   = val;
    labels[(size_t)img * KN + k] = j + 1;

    float local = (val > 0.f) ? fmaxf(fmaxf(x0, y0), fmaxf(x1, y1)) : 0.0f;
    atomicMax(&sMax, f2ord(local));
    __syncthreads();
    if (threadIdx.x == 0) atomicMax(&maxc[img], sMax);
}

// pairwise-IoU suppression bitmask; columns staged in LDS via async loads
__global__ void k_mask(const float* __restrict__ cb, const int* __restrict__ labels,
                       const unsigned int* __restrict__ maxc,
                       unsigned long long* __restrict__ mask) {
    __shared__ float sBox[64 * 4];
    __shared__ int   sLab[64];
    int img = blockIdx.z;
    int jw  = blockIdx.x;                       // column word [0,32)
    int t   = threadIdx.x;                      // 256 threads
    int i   = blockIdx.y * 256 + t;             // row
    float mc = ord2f(maxc[img]) + 1.0f;

    if (t < 64) {
        int j = jw * 64 + t;
        async_lds_b128(lds_addr(&sBox[t * 4]), cb + ((size_t)img * KN + j) * 4);
        async_lds_b32(lds_addr(&sLab[t]), labels + (size_t)img * KN + j);
        asm volatile("s_wait_asynccnt 0" ::: "memory");
    }
    __syncthreads();
    // apply per-class coordinate offset to staged columns (256 scalar entries)
    sBox[t] += (float)sLab[t >> 2] * mc;
    __syncthreads();

    const float* rb = cb + ((size_t)img * KN + i) * 4;
    float li = (float)labels[(size_t)img * KN + i] * mc;
    float ix0 = rb[0] + li, iy0 = rb[1] + li, ix1 = rb[2] + li, iy1 = rb[3] + li;
    float iarea = (ix1 - ix0) * (iy1 - iy0);

    unsigned long long m = 0ull;
    #pragma unroll 4
    for (int q = 0; q < 64; ++q) {
        float jx0 = sBox[q * 4 + 0], jy0 = sBox[q * 4 + 1];
        float jx1 = sBox[q * 4 + 2], jy1 = sBox[q * 4 + 3];
        float jarea = (jx1 - jx0) * (jy1 - jy0);
        float w = fmaxf(fminf(ix1, jx1) - fmaxf(ix0, jx0), 0.f);
        float h = fmaxf(fminf(iy1, jy1) - fmaxf(iy0, jy0), 0.f);
        float inter = w * h;
        float iou = inter / (iarea + jarea - inter + 1e-9f);
        if (iou > F_NMS_THRESH) m |= (1ull << q);
    }
    mask[((size_t)img * KN + i) * KW + jw] = m;
}

// serial greedy NMS scan; one wave32 per image.
// Mask rows staged through a double-buffered async global->LDS pipeline so the
// serial dependency chain only touches LDS + wave ballot (no global latency).
__global__ void k_scan(const unsigned long long* __restrict__ mask,
                       const float* __restrict__ vals,
                       unsigned long long* __restrict__ keep) {
    __shared__ unsigned long long buf[2][CH * KW];   // 2 x 8KB ping-pong
    int img = blockIdx.x;
    int w = threadIdx.x;                             // lane == column word [0,32)

    // per-lane cv bitmask for rows [w*64, w*64+64) (off the critical chain)
    const float* V = vals + (size_t)img * KN;
    unsigned long long cvw = 0ull;
    #pragma unroll 4
    for (int r = 0; r < 64; ++r)
        if (V[(w << 6) + r] > 0.f) cvw |= 1ull << r;

    const char* gbase = (const char*)(mask + (size_t)img * KN * KW);
    issue_chunk(&buf[0][0], gbase, w);               // prime pipeline

    unsigned long long keepw = 0ull;
    int base = w << 6;
    for (int c = 0; c < NCH; ++c) {
        if (c + 1 < NCH) {
            issue_chunk(&buf[(c + 1) & 1][0], gbase + (size_t)(c + 1) * CHUNK_BYTES, w);
            // async loads complete in order: <=16 outstanding => chunk c resident
            asm volatile("s_wait_asynccnt 16" ::: "memory");
        } else {
            asm volatile("s_wait_asynccnt 0" ::: "memory");
        }
        const unsigned long long* B = &buf[c & 1][0];
        #pragma unroll 4
        for (int r = 0; r < CH; ++r) {
            int i = (c << 5) + r;                    // global row
            unsigned long long row = B[r * KW + w];
            unsigned long long lm;
            if (i <= base)           lm = 0ull;
            else if (i >= base + 64) lm = ~0ull;
            else                     lm = (1ull << (i - base)) - 1ull;
            int sup = __any((row & keepw & lm) != 0ull);
            if (w == (c >> 1)) {                     // owner lane of word i>>6
                int bit = ((c & 1) << 5) + r;        // i & 63
                if (((cvw >> bit) & 1ull) && !sup) keepw |= 1ull << bit;
            }
        }
    }
    keep[(size_t)img * KW + w] = keepw;
}

// emit first 100 kept (order-preserving == final top-k), zero-fill the rest.
// Parallel: popcount + wave exclusive scan gives each kept entry its rank.
__global__ void k_final(const unsigned long long* __restrict__ keep,
                        const float* __restrict__ cb, const float* __restrict__ vals,
                        const int* __restrict__ labels, float* __restrict__ out) {
    int img = blockIdx.x, t = threadIdx.x;
    float* ob = out + (size_t)img * OUTN * 4;
    float* os = out + (size_t)BN * OUTN * 4 + (size_t)img * OUTN;
    float* ol = out + (size_t)BN * OUTN * 5 + (size_t)img * OUTN;
    for (int q = t; q < OUTN * 4; q += blockDim.x) ob[q] = 0.f;
    for (int q = t; q < OUTN;     q += blockDim.x) { os[q] = 0.f; ol[q] = 0.f; }
    __syncthreads();
    if (t < 32) {
        unsigned long long kw = keep[(size_t)img * KW + t];
        int cnt = __popcll(kw);
        int pre = cnt;                               // inclusive scan over lanes
        #pragma unroll
        for (int d = 1; d < 32; d <<= 1) {
            int other = __shfl_up(pre, d);
            if (t >= d) pre += other;
        }
        int rank = pre - cnt;                        // exclusive prefix = rank base
        unsigned long long word = kw;
        while (word != 0ull) {
            int b = __builtin_ctzll(word);
            if (rank < OUTN) {
                int i = (t << 6) + b;
                const float* bx = cb + ((size_t)img * KN + i) * 4;
                ob[rank * 4 + 0] = bx[0]; ob[rank * 4 + 1] = bx[1];
                ob[rank * 4 + 2] = bx[2]; ob[rank * 4 + 3] = bx[3];
                os[rank] = vals[(size_t)img * KN + i];
                ol[rank] = (float)labels[(size_t)img * KN + i];
            }
            ++rank;
            word &= word - 1ull;
        }
    }
}

extern "C" void kernel_launch(void* const* d_in, const int* in_sizes, int n_in,
                              void* d_out, int out_size, void* d_ws, size_t ws_size,
                              hipStream_t stream) {
    const float* logits = (const float*)d_in[0];   // (B,P,81)
    const float* deltas = (const float*)d_in[1];   // (B,P,324)
    const float* props  = (const float*)d_in[2];   // (B,P,4)

    char* ws = (char*)d_ws;
    float*              scores = (float*)(ws + WS_SCORES);
    unsigned int*       hist   = (unsigned int*)(ws + WS_HIST);
    unsigned long long* prefix = (unsigned long long*)(ws + WS_PREFIX);
    unsigned int*       remain = (unsigned int*)(ws + WS_REMAIN);
    unsigned int*       selcnt = (unsigned int*)(ws + WS_SELCNT);
    unsigned int*       maxc   = (unsigned int*)(ws + WS_MAXC);
    unsigned long long* keys   = (unsigned long long*)(ws + WS_KEYS);
    float*              cb     = (float*)(ws + WS_CB);
    float*              vals   = (float*)(ws + WS_VALS);
    int*                labels = (int*)(ws + WS_LAB);
    unsigned long long* mask   = (unsigned long long*)(ws + WS_MASK);
    unsigned long long* keep   = (unsigned long long*)(ws + WS_KEEP);

    k_init<<<BN, 256, 0, stream>>>(hist, prefix, remain, selcnt, maxc);
    k_scores<<<dim3((PN + 255) / 256, BN), 256, 0, stream>>>(logits, deltas, props, scores);
    for (int pass = 0; pass < 8; ++pass) {
        k_hist<<<dim3((PCN + 511) / 512, BN), 512, 0, stream>>>(scores, prefix, hist, pass);
        k_select<<<BN, 256, 0, stream>>>(hist, prefix, remain, pass);
    }
    k_compact<<<dim3((PCN + 511) / 512, BN), 512, 0, stream>>>(scores, prefix, selcnt, keys);
    k_sort<<<BN, 1024, 0, stream>>>(keys);
    k_decode<<<dim3(KN / 256, BN), 256, 0, stream>>>(keys, deltas, props, cb, vals, labels, maxc);
    k_mask<<<dim3(KW, KN / 256, BN), 256, 0, stream>>>(cb, labels, maxc, mask);
    k_scan<<<BN, 32, 0, stream>>>(mask, vals, keep);
    k_final<<<BN, 128, 0, stream>>>(keep, cb, vals, labels, (float*)d_out);
}